// GCN_47571057770576
// MI455X (gfx1250) — compile-verified
//
#include <hip/hip_runtime.h>

#define NN 50000
#define NE 800000

typedef float v2f __attribute__((ext_vector_type(2)));
typedef float v8f __attribute__((ext_vector_type(8)));

// ------------------------------------------------ degree counts (int)
__global__ __launch_bounds__(256) void k_zero_counts(int* outc, int* inc) {
  int i = blockIdx.x * 256 + threadIdx.x;
  if (i < NN) { outc[i] = 0; inc[i] = 0; }
}

__global__ __launch_bounds__(256) void k_count(const int* __restrict__ src,
                                               const int* __restrict__ dst,
                                               int* outc, int* inc) {
  int e = blockIdx.x * 256 + threadIdx.x;
  if (e < NE) {
    atomicAdd(&outc[src[e]], 1);
    atomicAdd(&inc[dst[e]], 1);
  }
}

__global__ __launch_bounds__(256) void k_norms(const int* __restrict__ outc,
                                               const int* __restrict__ inc,
                                               float* nsrc, float* ndst) {
  int i = blockIdx.x * 256 + threadIdx.x;
  if (i < NN) {                      // +1 = self-loop
    nsrc[i] = rsqrtf((float)(outc[i] + 1));
    ndst[i] = rsqrtf((float)(inc[i] + 1));
  }
}

// -------------------- single-workgroup exclusive scan of in-counts -> CSR offsets
__global__ __launch_bounds__(256) void k_scan(const int* __restrict__ cnt,
                                              int* __restrict__ off,
                                              int* __restrict__ cursor) {
  __shared__ int sh[256];
  __shared__ int carry;
  if (threadIdx.x == 0) carry = 0;
  __syncthreads();
  for (int base = 0; base < NN; base += 256) {
    int i = base + threadIdx.x;
    int v = (i < NN) ? cnt[i] : 0;
    sh[threadIdx.x] = v;
    __syncthreads();
    for (int s = 1; s < 256; s <<= 1) {       // Hillis-Steele inclusive scan
      int t = (threadIdx.x >= s) ? sh[threadIdx.x - s] : 0;
      __syncthreads();
      sh[threadIdx.x] += t;
      __syncthreads();
    }
    int incl = sh[threadIdx.x];
    int excl = incl - v;
    int c = carry;
    if (i < NN) { off[i] = c + excl; cursor[i] = c + excl; }
    __syncthreads();                          // all reads of carry done
    if (threadIdx.x == 255) carry = c + incl;
    __syncthreads();
  }
  if (threadIdx.x == 0) off[NN] = carry;      // == NE
}

// ------------------------------------------------ scatter edges into CSR
__global__ __launch_bounds__(256) void k_scatter(const int* __restrict__ src,
                                                 const int* __restrict__ dst,
                                                 int* __restrict__ cursor,
                                                 int* __restrict__ csrc) {
  int e = blockIdx.x * 256 + threadIdx.x;
  if (e < NE) {
    int p = atomicAdd(&cursor[dst[e]], 1);
    csrc[p] = src[e];
  }
}

// ------------------------------------------------ row scale: XS = X * nsrc
__global__ __launch_bounds__(256) void k_scale_rows(const float* __restrict__ X,
                                                    const float* __restrict__ nsrc,
                                                    float* __restrict__ XS,
                                                    int dq, int d) {
  long long t = (long long)blockIdx.x * 256 + threadIdx.x;
  long long total = (long long)NN * dq;
  if (t >= total) return;
  int row = (int)(t / dq);
  int cq  = (int)(t % dq);
  float s = nsrc[row];
  float4 x = ((const float4*)(X + (long long)row * d))[cq];
  float4 y; y.x = x.x * s; y.y = x.y * s; y.z = x.z * s; y.w = x.w * s;
  ((float4*)(XS + (long long)row * d))[cq] = y;
}

// ------------------- CSR gather-accumulate: M[dst] = XS[dst] + sum XS[src]
// No float atomics: register accumulation, one store per output chunk.
// Gathers are L2-resident (51 MB feature matrix << 192 MB L2).
__global__ __launch_bounds__(256) void k_agg_csr(const float* __restrict__ XS,
                                                 const int* __restrict__ off,
                                                 const int* __restrict__ csrc,
                                                 float* __restrict__ M,
                                                 int dq, int d) {
  long long t = (long long)blockIdx.x * 256 + threadIdx.x;
  long long total = (long long)NN * dq;
  if (t >= total) return;
  int row = (int)(t / dq);
  int cq  = (int)(t % dq);
  float4 acc = ((const float4*)(XS + (long long)row * d))[cq];  // self-loop
  int e0 = off[row], e1 = off[row + 1];
  for (int e = e0; e < e1; ++e) {
    int s = csrc[e];                                            // wave-uniform
    float4 x = ((const float4*)(XS + (long long)s * d))[cq];
    acc.x += x.x; acc.y += x.y; acc.z += x.z; acc.w += x.w;
  }
  ((float4*)(M + (long long)row * d))[cq] = acc;
}

// -------- fp32 WMMA GEMM, 16x64 tile per wave: Y = act((M .* ndst) @ W + b)
// One A fragment feeds 4 V_WMMA_F32_16X16X4_F32 per K-step (4x A reuse).
__global__ __launch_bounds__(256) void k_gemm_wmma(const float* __restrict__ A,
                                                   const float* __restrict__ ndst,
                                                   const float* __restrict__ W,
                                                   const float* __restrict__ bias,
                                                   float* __restrict__ Y,
                                                   int din, int dout, int tilesN64,
                                                   int relu) {
  int wid  = (blockIdx.x * 256 + threadIdx.x) >> 5;  // global wave id
  int lane = threadIdx.x & 31;
  int tN = wid % tilesN64;
  int tM = wid / tilesN64;
  if (tM >= NN / 16) return;                          // wave-uniform guard
  int li = lane & 15, hi = lane >> 4;
  int rowA = tM * 16 + li;
  int col0 = tN * 64 + li;
  float nd = ndst[rowA];
  const float* Ar = A + (long long)rowA * din;
  v8f acc0 = (v8f){0.f,0.f,0.f,0.f,0.f,0.f,0.f,0.f};
  v8f acc1 = acc0, acc2 = acc0, acc3 = acc0;
#pragma unroll 2
  for (int k0 = 0; k0 < din; k0 += 4) {
    int kA = k0 + hi * 2;
    v2f a;
    a.x = Ar[kA]     * nd;           // fuse norm_dst row scale into A load
    a.y = Ar[kA + 1] * nd;
    const float* Wr0 = W + (long long)kA * dout;
    const float* Wr1 = W + (long long)(kA + 1) * dout;
    v2f b0, b1, b2, b3;
    b0.x = Wr0[col0];      b0.y = Wr1[col0];
    b1.x = Wr0[col0 + 16]; b1.y = Wr1[col0 + 16];
    b2.x = Wr0[col0 + 32]; b2.y = Wr1[col0 + 32];
    b3.x = Wr0[col0 + 48]; b3.y = Wr1[col0 + 48];
    acc0 = __builtin_amdgcn_wmma_f32_16x16x4_f32(false, a, false, b0, (short)0, acc0, false, false);
    acc1 = __builtin_amdgcn_wmma_f32_16x16x4_f32(false, a, false, b1, (short)0, acc1, false, false);
    acc2 = __builtin_amdgcn_wmma_f32_16x16x4_f32(false, a, false, b2, (short)0, acc2, false, false);
    acc3 = __builtin_amdgcn_wmma_f32_16x16x4_f32(false, a, false, b3, (short)0, acc3, false, false);
  }
  long long rbase = (long long)(tM * 16 + hi * 8) * dout;
#pragma unroll
  for (int g = 0; g < 4; ++g) {
    v8f acc = (g == 0) ? acc0 : (g == 1) ? acc1 : (g == 2) ? acc2 : acc3;
    int col = col0 + g * 16;
    float bv = bias[col];
#pragma unroll
    for (int r = 0; r < 8; r++) {
      float v = acc[r] + bv;
      if (relu) v = fmaxf(v, 0.0f);
      Y[rbase + (long long)r * dout + col] = v;
    }
  }
}

// ---------------------------------------------------------------- launcher
extern "C" void kernel_launch(void* const* d_in, const int* in_sizes, int n_in,
                              void* d_out, int out_size, void* d_ws, size_t ws_size,
                              hipStream_t stream) {
  const float* feats = (const float*)d_in[0];
  const float* W1 = (const float*)d_in[1];
  const float* b1 = (const float*)d_in[2];
  const float* W2 = (const float*)d_in[3];
  const float* b2 = (const float*)d_in[4];
  const float* W3 = (const float*)d_in[5];
  const float* b3 = (const float*)d_in[6];
  const int* src  = (const int*)d_in[7];
  const int* dst  = (const int*)d_in[8];
  float* out = (float*)d_out;

  float* ws   = (float*)d_ws;
  float* nsrc = ws;                        // [NN]
  float* ndst = ws + NN;                   // [NN]
  float* XS   = ws + 2 * NN;               // [NN*256]
  float* Mb   = XS + (size_t)NN * 256;     // [NN*256]
  float* Hb   = Mb + (size_t)NN * 256;     // [NN*256]
  int* ip     = (int*)(Hb + (size_t)NN * 256);
  int* outc   = ip;                        // [NN]   (reused: only for norms)
  int* inc    = ip + NN;                   // [NN]
  int* rowoff = ip + 2 * NN;               // [NN+1]
  int* cursor = rowoff + (NN + 1);         // [NN]
  int* csrc   = cursor + NN;               // [NE]

  auto cdiv = [](long long a, long long b) { return (int)((a + b - 1) / b); };

  // degrees -> norms, CSR build
  k_zero_counts<<<cdiv(NN, 256), 256, 0, stream>>>(outc, inc);
  k_count<<<cdiv(NE, 256), 256, 0, stream>>>(src, dst, outc, inc);
  k_norms<<<cdiv(NN, 256), 256, 0, stream>>>(outc, inc, nsrc, ndst);
  k_scan<<<1, 256, 0, stream>>>(inc, rowoff, cursor);
  k_scatter<<<cdiv(NE, 256), 256, 0, stream>>>(src, dst, cursor, csrc);

  const int tilesM = NN / 16;  // 3125, exact

  // layer 1: 128 -> 256, relu
  k_scale_rows<<<cdiv((long long)NN * 32, 256), 256, 0, stream>>>(feats, nsrc, XS, 32, 128);
  k_agg_csr<<<cdiv((long long)NN * 32, 256), 256, 0, stream>>>(XS, rowoff, csrc, Mb, 32, 128);
  k_gemm_wmma<<<cdiv((long long)tilesM * 4, 8), 256, 0, stream>>>(Mb, ndst, W1, b1, Hb, 128, 256, 4, 1);

  // layer 2: 256 -> 256, relu
  k_scale_rows<<<cdiv((long long)NN * 64, 256), 256, 0, stream>>>(Hb, nsrc, XS, 64, 256);
  k_agg_csr<<<cdiv((long long)NN * 64, 256), 256, 0, stream>>>(XS, rowoff, csrc, Mb, 64, 256);
  k_gemm_wmma<<<cdiv((long long)tilesM * 4, 8), 256, 0, stream>>>(Mb, ndst, W2, b2, Hb, 256, 256, 4, 1);

  // layer 3: 256 -> 64, no act, writes d_out
  k_scale_rows<<<cdiv((long long)NN * 64, 256), 256, 0, stream>>>(Hb, nsrc, XS, 64, 256);
  k_agg_csr<<<cdiv((long long)NN * 64, 256), 256, 0, stream>>>(XS, rowoff, csrc, Mb, 64, 256);
  k_gemm_wmma<<<cdiv((long long)tilesM * 1, 8), 256, 0, stream>>>(Mb, ndst, W3, b3, out, 256, 64, 1, 0);
}